// RandomizedWeightedProcrustes_2173253452313
// MI455X (gfx1250) — compile-verified
//
#include <hip/hip_runtime.h>
#include <math.h>

#define BB 4
#define KK 256
#define SS 256
#define PPT 4
#define TILE (256 * PPT)

typedef float v2f __attribute__((ext_vector_type(2)));
typedef float v8f __attribute__((ext_vector_type(8)));

__device__ inline float wave_sum(float v) {
#pragma unroll
  for (int off = 16; off > 0; off >>= 1) v += __shfl_xor(v, off, 32);
  return v;
}

// One wave (32 lanes) per hypothesis (b,k):
//  pass 1 : gather S=256 samples, weighted sums -> centroids, stage raw data in LDS
//  pass 1b: rewrite LDS in place: s_src = src0 (centered), s_tgt = wn * tgt0
//  pass 2 : covariance H (3x3 padded into 16x16) via chained V_WMMA_F32_16X16X4_F32
//           (branchless: clamped LDS address + 0/1 lane mask, EXEC stays all-ones)
//  then Horn quaternion -> R, t -> store 4x4 transform
__global__ __launch_bounds__(32) void k_procrustes(
    const float* __restrict__ src, const float* __restrict__ tgt,
    const float* __restrict__ w, const int* __restrict__ sel,
    int N, float* __restrict__ transforms, float* __restrict__ errors) {
  __shared__ float s_src[SS * 3];
  __shared__ float s_tgt[SS * 3];
  __shared__ float s_w[SS];

  const int bk = blockIdx.x;
  const int b = bk >> 8;        // / KK
  const int k = bk & (KK - 1);  // % KK
  const int lane = threadIdx.x;

  const float* srcB = src + (size_t)b * N * 3;
  const float* tgtB = tgt + (size_t)b * N * 3;
  const float* wB = w + (size_t)b * N;
  const int* selK = sel + (size_t)k * SS;

  float sw = 0.f, sx = 0.f, sy = 0.f, sz = 0.f, tx = 0.f, ty = 0.f, tz = 0.f;
#pragma unroll
  for (int c = 0; c < SS / 32; ++c) {
    int s = c * 32 + lane;
    int idx = selK[s];
    float wv = wB[idx];
    float a0 = srcB[idx * 3 + 0], a1 = srcB[idx * 3 + 1], a2 = srcB[idx * 3 + 2];
    float b0 = tgtB[idx * 3 + 0], b1 = tgtB[idx * 3 + 1], b2 = tgtB[idx * 3 + 2];
    s_src[s * 3 + 0] = a0; s_src[s * 3 + 1] = a1; s_src[s * 3 + 2] = a2;
    s_tgt[s * 3 + 0] = b0; s_tgt[s * 3 + 1] = b1; s_tgt[s * 3 + 2] = b2;
    s_w[s] = wv;
    sw += wv;
    sx += wv * a0; sy += wv * a1; sz += wv * a2;
    tx += wv * b0; ty += wv * b1; tz += wv * b2;
  }
  sw = wave_sum(sw);
  sx = wave_sum(sx); sy = wave_sum(sy); sz = wave_sum(sz);
  tx = wave_sum(tx); ty = wave_sum(ty); tz = wave_sum(tz);
  const float winv = 1.0f / (sw + 1e-5f);
  const float scx = sx * winv, scy = sy * winv, scz = sz * winv;
  const float tcx = tx * winv, tcy = ty * winv, tcz = tz * winv;
  __syncthreads();

  // Pass 1b: center src, fold normalized weight into tgt. In-place in LDS.
#pragma unroll
  for (int c = 0; c < SS / 32; ++c) {
    int s = c * 32 + lane;
    float wn = s_w[s] * winv;
    s_src[s * 3 + 0] -= scx;
    s_src[s * 3 + 1] -= scy;
    s_src[s * 3 + 2] -= scz;
    s_tgt[s * 3 + 0] = wn * (s_tgt[s * 3 + 0] - tcx);
    s_tgt[s * 3 + 1] = wn * (s_tgt[s * 3 + 1] - tcy);
    s_tgt[s * 3 + 2] = wn * (s_tgt[s * 3 + 2] - tcz);
  }
  __syncthreads();

  // WMMA covariance. A-layout (16x4 f32): lane l holds A[m=l&15][k=(l>>4)*2 + {0,1}]
  // B-layout (4x16 f32): lane l holds B[(l>>4)*2 + {0,1}][n=l&15]
  // Branchless: lanes with m/n >= 3 read a clamped (valid) address and mask to 0.
  const int mn = lane & 15;
  const int kh = (lane >> 4) * 2;
  const int mnc = (mn < 3) ? mn : 0;
  const float msk = (mn < 3) ? 1.f : 0.f;

  v8f acc = {};
  for (int g = 0; g < SS / 4; ++g) {
    const int s0 = g * 4 + kh;
    v2f av, bv;
    av[0] = msk * s_src[s0 * 3 + mnc];
    av[1] = msk * s_src[s0 * 3 + 3 + mnc];
    bv[0] = msk * s_tgt[s0 * 3 + mnc];
    bv[1] = msk * s_tgt[s0 * 3 + 3 + mnc];
    acc = __builtin_amdgcn_wmma_f32_16x16x4_f32(false, av, false, bv,
                                                (short)0, acc, false, false);
  }

  // H[i][j] = acc[i] at lane j (C layout: vgpr r, lanes 0-15 -> M=r, N=lane)
  float c0 = acc[0], c1 = acc[1], c2 = acc[2];
  float Sxx = __shfl(c0, 0, 32), Sxy = __shfl(c0, 1, 32), Sxz = __shfl(c0, 2, 32);
  float Syx = __shfl(c1, 0, 32), Syy = __shfl(c1, 1, 32), Syz = __shfl(c1, 2, 32);
  float Szx = __shfl(c2, 0, 32), Szy = __shfl(c2, 1, 32), Szz = __shfl(c2, 2, 32);

  // Horn's quaternion method: max eigenvector of symmetric 4x4 N.
  float N00 = Sxx + Syy + Szz;
  float N01 = Syz - Szy;
  float N02 = Szx - Sxz;
  float N03 = Sxy - Syx;
  float N11 = Sxx - Syy - Szz;
  float N12 = Sxy + Syx;
  float N13 = Szx + Sxz;
  float N22 = -Sxx + Syy - Szz;
  float N23 = Syz + Szy;
  float N33 = -Sxx - Syy + Szz;
  float sh = fabsf(N00) + fabsf(N11) + fabsf(N22) + fabsf(N33) +
             2.f * (fabsf(N01) + fabsf(N02) + fabsf(N03) + fabsf(N12) +
                    fabsf(N13) + fabsf(N23)) +
             1e-9f;
  float qw = 1.f, qx = 0.f, qy = 0.f, qz = 0.f;
#pragma unroll 1
  for (int it = 0; it < 40; ++it) {
    float nw = N00 * qw + N01 * qx + N02 * qy + N03 * qz + sh * qw;
    float nx = N01 * qw + N11 * qx + N12 * qy + N13 * qz + sh * qx;
    float ny = N02 * qw + N12 * qx + N22 * qy + N23 * qz + sh * qy;
    float nz = N03 * qw + N13 * qx + N23 * qy + N33 * qz + sh * qz;
    float inv = rsqrtf(nw * nw + nx * nx + ny * ny + nz * nz + 1e-30f);
    qw = nw * inv; qx = nx * inv; qy = ny * inv; qz = nz * inv;
  }
  float r00 = 1.f - 2.f * (qy * qy + qz * qz);
  float r01 = 2.f * (qx * qy - qw * qz);
  float r02 = 2.f * (qx * qz + qw * qy);
  float r10 = 2.f * (qx * qy + qw * qz);
  float r11 = 1.f - 2.f * (qx * qx + qz * qz);
  float r12 = 2.f * (qy * qz - qw * qx);
  float r20 = 2.f * (qx * qz - qw * qy);
  float r21 = 2.f * (qy * qz + qw * qx);
  float r22 = 1.f - 2.f * (qx * qx + qy * qy);
  float t0 = tcx - (r00 * scx + r01 * scy + r02 * scz);
  float t1 = tcy - (r10 * scx + r11 * scy + r12 * scz);
  float t2 = tcz - (r20 * scx + r21 * scy + r22 * scz);

  if (lane < 16) {
    int row = lane >> 2, col = lane & 3;
    float val;
    if (row == 0)      val = (col == 0) ? r00 : ((col == 1) ? r01 : ((col == 2) ? r02 : t0));
    else if (row == 1) val = (col == 0) ? r10 : ((col == 1) ? r11 : ((col == 2) ? r12 : t1));
    else if (row == 2) val = (col == 0) ? r20 : ((col == 1) ? r21 : ((col == 2) ? r22 : t2));
    else               val = (col == 3) ? 1.f : 0.f;
    transforms[(size_t)bk * 16 + lane] = val;
  }
  if (lane == 0) errors[bk] = 0.f;  // zero accumulator for the scoring pass
}

// Tile over points; loop all K hypotheses with R,t staged in LDS.
// Per-wave reduction -> LDS float atomics -> one global atomic per (block,k).
__global__ __launch_bounds__(256) void k_score(
    const float* __restrict__ src, const float* __restrict__ tgt,
    const float* __restrict__ w, const float* __restrict__ transforms,
    int N, float* __restrict__ errors) {
  __shared__ float sRt[KK * 12];
  __shared__ float s_err[KK];
  const int b = blockIdx.y;
  const int tid = threadIdx.x;
  {
    const float* T = transforms + (size_t)(b * KK + tid) * 16;
#pragma unroll
    for (int i = 0; i < 12; ++i) sRt[tid * 12 + i] = T[i];
    s_err[tid] = 0.f;
  }
  __syncthreads();

  const float* sB = src + (size_t)b * N * 3;
  const float* tB = tgt + (size_t)b * N * 3;
  const float* wB = w + (size_t)b * N;
  const int n0 = blockIdx.x * TILE;

  float px[PPT], py[PPT], pz[PPT], gx[PPT], gy[PPT], gz[PPT], pw[PPT];
#pragma unroll
  for (int i = 0; i < PPT; ++i) {
    int n = n0 + tid + i * 256;
    bool ok = (n < N);
    int nn = ok ? n : 0;
    px[i] = sB[nn * 3 + 0]; py[i] = sB[nn * 3 + 1]; pz[i] = sB[nn * 3 + 2];
    gx[i] = tB[nn * 3 + 0]; gy[i] = tB[nn * 3 + 1]; gz[i] = tB[nn * 3 + 2];
    pw[i] = ok ? wB[nn] : 0.f;
  }

  for (int k = 0; k < KK; ++k) {
    const float* Rt = &sRt[k * 12];
    float R00 = Rt[0], R01 = Rt[1], R02 = Rt[2], T0 = Rt[3];
    float R10 = Rt[4], R11 = Rt[5], R12 = Rt[6], T1 = Rt[7];
    float R20 = Rt[8], R21 = Rt[9], R22 = Rt[10], T2 = Rt[11];
    float accu = 0.f;
#pragma unroll
    for (int i = 0; i < PPT; ++i) {
      float ax = R00 * px[i] + R01 * py[i] + R02 * pz[i] + T0 - gx[i];
      float ay = R10 * px[i] + R11 * py[i] + R12 * pz[i] + T1 - gy[i];
      float az = R20 * px[i] + R21 * py[i] + R22 * pz[i] + T2 - gz[i];
      accu += sqrtf(ax * ax + ay * ay + az * az) * pw[i];
    }
    accu = wave_sum(accu);
    if ((tid & 31) == 0) atomicAdd(&s_err[k], accu);  // ds_add_f32
  }
  __syncthreads();
  atomicAdd(&errors[b * KK + tid], s_err[tid]);  // one global atomic per (block,k)
}

// Per-batch argmin over K and copy best 4x4 transform to output.
__global__ __launch_bounds__(256) void k_select(
    const float* __restrict__ errors, const float* __restrict__ transforms,
    float* __restrict__ out) {
  __shared__ float sv[256];
  __shared__ int si[256];
  const int b = blockIdx.x;
  const int tid = threadIdx.x;
  sv[tid] = errors[b * KK + tid];
  si[tid] = tid;
  __syncthreads();
  for (int off = 128; off > 0; off >>= 1) {
    if (tid < off) {
      float v2 = sv[tid + off];
      int i2 = si[tid + off];
      if (v2 < sv[tid] || (v2 == sv[tid] && i2 < si[tid])) {
        sv[tid] = v2;
        si[tid] = i2;
      }
    }
    __syncthreads();
  }
  int best = si[0];
  if (tid < 16) out[b * 16 + tid] = transforms[(size_t)(b * KK + best) * 16 + tid];
}

extern "C" void kernel_launch(void* const* d_in, const int* in_sizes, int n_in,
                              void* d_out, int out_size, void* d_ws, size_t ws_size,
                              hipStream_t stream) {
  const float* src = (const float*)d_in[0];  // (B,N,3)
  const float* tgt = (const float*)d_in[1];  // (B,N,3)
  const float* w = (const float*)d_in[2];    // (B,N)
  const int* sel = (const int*)d_in[3];      // (K,S) int32
  float* out = (float*)d_out;                // (B,4,4)
  const int N = in_sizes[2] / BB;

  float* transforms = (float*)d_ws;           // BB*KK*16 floats
  float* errors = transforms + BB * KK * 16;  // BB*KK floats

  k_procrustes<<<BB * KK, 32, 0, stream>>>(src, tgt, w, sel, N, transforms, errors);
  int tiles = (N + TILE - 1) / TILE;
  k_score<<<dim3(tiles, BB), 256, 0, stream>>>(src, tgt, w, transforms, N, errors);
  k_select<<<BB, 256, 0, stream>>>(errors, transforms, out);
}